// GCN_Encoder_61770219651564
// MI455X (gfx1250) — compile-verified
//
#include <hip/hip_runtime.h>
#include <hip/hip_bf16.h>

typedef __attribute__((ext_vector_type(2))) float v2f;
typedef __attribute__((ext_vector_type(8))) float v8f;

#define G_GRAPHS 64
#define EMB_F    128

static __host__ __device__ inline int ceil_div_i(int a, int b) { return (a + b - 1) / b; }

// ---------------------------------------------------------------------------
// Degree / normalization
// ---------------------------------------------------------------------------
__global__ void k_deg_init(float* __restrict__ deg, int N) {
    int i = blockIdx.x * blockDim.x + threadIdx.x;
    if (i < N) deg[i] = 1.0f;  // self-loop
}

__global__ void k_deg_count(const int* __restrict__ dst, float* __restrict__ deg, int E) {
    int e = blockIdx.x * blockDim.x + threadIdx.x;
    if (e < E) atomicAdd(&deg[dst[e]], 1.0f);
}

__global__ void k_dinv(float* __restrict__ deg, int N) {
    int i = blockIdx.x * blockDim.x + threadIdx.x;
    if (i < N) {
        float d = fmaxf(deg[i], 1.0f);
        deg[i] = 1.0f / sqrtf(d);  // in-place: deg buffer becomes dinv
    }
}

// ---------------------------------------------------------------------------
// Dense transform T = X @ W  via V_WMMA_F32_16X16X4_F32.
//   * One 256-thread block = 8 waves; W staged once into LDS in a
//     K-pair-interleaved layout so each lane's B fragment is one aligned
//     8-byte word:  sW[(k>>1)*2*MOUT + 2*col + (k&1)] = W[k][col]
//     -> single ds_load_b64 straight into an even-aligned VGPR pair.
//   * Each wave computes a full 16 x MOUT row-stripe: the A fragment is
//     loaded once per K-step (global_load_b64) and reused across MOUT/16
//     back-to-back WMMAs.
//   X: [N,K] row-major, W: [K,MOUT] row-major, T: [N,MOUT] row-major.
//   Requires N % 16 == 0 (true: 100000 = 6250*16).
// ---------------------------------------------------------------------------
template <int K, int MOUT>
__global__ __launch_bounds__(256) void k_gemm_wmma(const float* __restrict__ X,
                                                   const float* __restrict__ W,
                                                   float* __restrict__ T, int N) {
    constexpr int TILES = MOUT / 16;
    __shared__ float sW[K * MOUT];                     // <= 64 KB
    // stage W with K-pair interleave: pair p=k>>1, lane word = {W[2p][c], W[2p+1][c]}
    for (int i = threadIdx.x; i < K * MOUT; i += 256) {
        int k = i / MOUT;
        int c = i - k * MOUT;
        sW[(k >> 1) * (2 * MOUT) + (c << 1) + (k & 1)] = W[i];
    }
    __syncthreads();

    const int waveId = threadIdx.x >> 5;
    const int lane   = threadIdx.x & 31;
    const int row0   = (blockIdx.x * 8 + waveId) << 4;
    if (row0 >= N) return;                             // wave-uniform

    const int half = lane >> 4;    // 0: K=k..k+1 slice, 1: K=k+2..k+3 slice
    const int l16  = lane & 15;

    const float* __restrict__ Xrow = X + (size_t)(row0 + l16) * K;

    v8f acc[TILES];
    const v8f zero = {0.f, 0.f, 0.f, 0.f, 0.f, 0.f, 0.f, 0.f};
#pragma unroll
    for (int t = 0; t < TILES; ++t) acc[t] = zero;

    for (int k = 0; k < K; k += 4) {
        const int ka = k + (half << 1);               // this lane's K pair base
        v2f a;
        a.x = Xrow[ka + 0];
        a.y = Xrow[ka + 1];
        // LDS base for this lane's K pair (pair index = ka>>1)
        const float* __restrict__ sp = sW + (ka >> 1) * (2 * MOUT) + (l16 << 1);
#pragma unroll
        for (int t = 0; t < TILES; ++t) {
            v2f b = *reinterpret_cast<const v2f*>(sp + t * 32);  // ds_load_b64
            acc[t] = __builtin_amdgcn_wmma_f32_16x16x4_f32(
                         /*neg_a=*/false, a, /*neg_b=*/false, b,
                         /*c_mod=*/(short)0, acc[t],
                         /*reuse_a=*/false, /*reuse_b=*/false);
        }
    }

    // C/D layout: VGPR j holds row (row0 + 8*half + j), col = tile*16 + l16
#pragma unroll
    for (int t = 0; t < TILES; ++t) {
        float* __restrict__ Tp =
            T + (size_t)(row0 + (half << 3)) * MOUT + t * 16 + l16;
#pragma unroll
        for (int j = 0; j < 8; ++j) Tp[(size_t)j * MOUT] = acc[t][j];
    }
}

// ---------------------------------------------------------------------------
// Aggregation: agg[v] = dinv[v]^2 * T[v]   (self-loop term; also inits buffer)
// ---------------------------------------------------------------------------
__global__ void k_self_init(const float* __restrict__ T,
                            const float* __restrict__ dinv,
                            float* __restrict__ agg, int N, int F) {
    int idx = blockIdx.x * blockDim.x + threadIdx.x;    // over N * F/4
    const int chunks = F >> 2;
    if (idx >= N * chunks) return;
    int v = idx / chunks;
    float w = dinv[v];
    float w2 = w * w;
    float4 m = reinterpret_cast<const float4*>(T)[idx];
    float4 r = make_float4(w2 * m.x, w2 * m.y, w2 * m.z, w2 * m.w);
    reinterpret_cast<float4*>(agg)[idx] = r;
}

// agg[dst] += dinv[src]*dinv[dst] * T[src]; one thread per (edge, 4 features)
__global__ void k_edge_scatter(const int* __restrict__ src,
                               const int* __restrict__ dst,
                               const float* __restrict__ dinv,
                               const float* __restrict__ T,
                               float* __restrict__ agg, int E, int F) {
    int t = blockIdx.x * blockDim.x + threadIdx.x;
    const int chunks = F >> 2;
    if (t >= E * chunks) return;
    int e  = t / chunks;
    int f4 = (t - e * chunks) << 2;
    int s = src[e];
    int d = dst[e];
    float w = dinv[s] * dinv[d];
    const float4 m = *reinterpret_cast<const float4*>(T + (size_t)s * F + f4);
    float* __restrict__ ap = agg + (size_t)d * F + f4;
    atomicAdd(ap + 0, w * m.x);
    atomicAdd(ap + 1, w * m.y);
    atomicAdd(ap + 2, w * m.z);
    atomicAdd(ap + 3, w * m.w);
}

__global__ void k_finalize(const float* __restrict__ agg,
                           const float* __restrict__ bias,
                           float* __restrict__ H, int N, int F, int relu) {
    int idx = blockIdx.x * blockDim.x + threadIdx.x;    // over N * F/4
    const int chunks = F >> 2;
    if (idx >= N * chunks) return;
    int f4 = (idx % chunks) << 2;
    float4 a = reinterpret_cast<const float4*>(agg)[idx];
    float4 b = *reinterpret_cast<const float4*>(bias + f4);
    float4 r = make_float4(a.x + b.x, a.y + b.y, a.z + b.z, a.w + b.w);
    if (relu) {
        r.x = fmaxf(r.x, 0.f); r.y = fmaxf(r.y, 0.f);
        r.z = fmaxf(r.z, 0.f); r.w = fmaxf(r.w, 0.f);
    }
    reinterpret_cast<float4*>(H)[idx] = r;
}

// ---------------------------------------------------------------------------
// Global mean pool
// ---------------------------------------------------------------------------
__global__ void k_pool_init(float* __restrict__ sums, float* __restrict__ cnt) {
    int i = blockIdx.x * blockDim.x + threadIdx.x;
    if (i < G_GRAPHS * EMB_F) sums[i] = 0.0f;
    if (i < G_GRAPHS) cnt[i] = 0.0f;
}

__global__ void k_pool_acc(const float* __restrict__ H,
                           const int* __restrict__ batch,
                           float* __restrict__ sums,
                           float* __restrict__ cnt, int N, int F) {
    int idx = blockIdx.x * blockDim.x + threadIdx.x;
    if (idx >= N * F) return;
    int v = idx / F;
    int f = idx - v * F;
    int g = batch[v];
    atomicAdd(&sums[g * F + f], H[idx]);
    if (f == 0) atomicAdd(&cnt[g], 1.0f);
}

__global__ void k_pool_div(const float* __restrict__ sums,
                           const float* __restrict__ cnt,
                           float* __restrict__ out) {
    int idx = blockIdx.x * blockDim.x + threadIdx.x;
    if (idx >= G_GRAPHS * EMB_F) return;
    int g = idx / EMB_F;
    out[idx] = sums[idx] / fmaxf(cnt[g], 1.0f);
}

// ---------------------------------------------------------------------------
// Launch
// ---------------------------------------------------------------------------
extern "C" void kernel_launch(void* const* d_in, const int* in_sizes, int n_in,
                              void* d_out, int out_size, void* d_ws, size_t ws_size,
                              hipStream_t stream) {
    const float* x     = (const float*)d_in[0];
    const int*   ei    = (const int*)  d_in[1];
    const int*   batch = (const int*)  d_in[2];
    const float* W1    = (const float*)d_in[3];
    const float* b1    = (const float*)d_in[4];
    const float* W2    = (const float*)d_in[5];
    const float* b2    = (const float*)d_in[6];
    const float* W3    = (const float*)d_in[7];
    const float* b3    = (const float*)d_in[8];
    float*       out   = (float*)d_out;

    const int E = in_sizes[1] / 2;
    const int N = in_sizes[0] / 128;
    const int D_IN = 128, HID = 64, HID2 = 128, EMB = 128;

    const int* src = ei;
    const int* dst = ei + E;

    float* ws   = (float*)d_ws;
    float* dinv = ws;                               // N floats (deg, then dinv)
    float* A    = ws + N;                           // N*128 (transformed feats)
    float* B    = A + (size_t)N * 128;              // N*128 (aggregation acc)
    float* C    = B + (size_t)N * 128;              // N*128 (layer activations)
    float* sums = C + (size_t)N * 128;              // 64*128
    float* cnt  = sums + G_GRAPHS * EMB_F;          // 64

    const int TB = 256;
    const int gemm_blocks = ceil_div_i(N, 16 * 8);  // 8 row-stripes per block

    // --- degree / dinv ---
    k_deg_init <<<ceil_div_i(N, TB), TB, 0, stream>>>(dinv, N);
    k_deg_count<<<ceil_div_i(E, TB), TB, 0, stream>>>(dst, dinv, E);
    k_dinv     <<<ceil_div_i(N, TB), TB, 0, stream>>>(dinv, N);

    // --- one GCN conv layer: message passing around the WMMA transform ---
    auto post = [&](const float* bias, int F, int relu) {
        k_self_init   <<<ceil_div_i(N * (F / 4), TB), TB, 0, stream>>>(A, dinv, B, N, F);
        k_edge_scatter<<<ceil_div_i(E * (F / 4), TB), TB, 0, stream>>>(src, dst, dinv, A, B, E, F);
        k_finalize    <<<ceil_div_i(N * (F / 4), TB), TB, 0, stream>>>(B, bias, C, N, F, relu);
    };

    // conv1: 128 -> 64, ReLU
    k_gemm_wmma<128, 64><<<gemm_blocks, TB, 0, stream>>>(x, W1, A, N);
    post(b1, HID, 1);
    // conv2: 64 -> 128, ReLU
    k_gemm_wmma<64, 128><<<gemm_blocks, TB, 0, stream>>>(C, W2, A, N);
    post(b2, HID2, 1);
    // conv3: 128 -> 128
    k_gemm_wmma<128, 128><<<gemm_blocks, TB, 0, stream>>>(C, W3, A, N);
    post(b3, EMB, 0);

    // --- global mean pool ---
    k_pool_init<<<ceil_div_i(G_GRAPHS * EMB_F, TB), TB, 0, stream>>>(sums, cnt);
    k_pool_acc <<<ceil_div_i(N * EMB_F, TB), TB, 0, stream>>>(C, batch, sums, cnt, N, EMB_F);
    k_pool_div <<<ceil_div_i(G_GRAPHS * EMB_F, TB), TB, 0, stream>>>(sums, cnt, out);
}